// RegularGridInterpolator_9131100471569
// MI455X (gfx1250) — compile-verified
//
#include <hip/hip_runtime.h>
#include <stdint.h>

#define GRID_N 256

typedef __attribute__((ext_vector_type(2))) float f32x2;

// Byte offset of a __shared__ object within the wave's LDS allocation
// (generic -> addrspace(3) cast, then ptrtoint of the 32-bit LDS pointer).
__device__ __forceinline__ unsigned lds_byte_off(const void* p) {
    return (unsigned)(unsigned long long)
        (__attribute__((address_space(3))) const void*)p;
}

struct AxisTerm {
    unsigned il, ir;   // left/right bracket indices
    float    wl, wr;   // weight applied to left / right corner (= opposite distance)
    float    ov;       // dist_left + dist_right (denominator factor)
};

// Matches jnp.searchsorted(p, c, side='left') + clamp + distance edge cases.
__device__ __forceinline__ AxisTerm axis_terms(const float* __restrict__ s, float c) {
    // branchless lower_bound over 256 sorted LDS floats (8 dependent ds_loads)
    unsigned pos = 0u;
#pragma unroll
    for (unsigned step = 128u; step >= 1u; step >>= 1u) {
        if (s[pos + step - 1u] < c) pos += step;
    }
    unsigned ir = pos < 255u ? pos : 255u;        // min(idx_right, n-1)
    unsigned il = ir > 0u ? ir - 1u : 0u;         // clip(idx_right-1, 0, n-1)
    float dl = fmaxf(c - s[il], 0.0f);
    float dr = fmaxf(s[ir] - c, 0.0f);
    if (dl == 0.0f && dr == 0.0f) { dl = 1.0f; dr = 1.0f; }
    AxisTerm t;
    t.il = il; t.ir = ir;
    t.wl = dr;            // left corner weighted by right distance
    t.wr = dl;            // right corner weighted by left distance
    t.ov = dl + dr;
    return t;
}

// Full trilinear inverse-distance blend for one query point.
__device__ __forceinline__ float interp_one(const float* __restrict__ sax,
                                            const float* __restrict__ values,
                                            float cx, float cy, float cz)
{
    const AxisTerm ax = axis_terms(&sax[0],          cx);
    const AxisTerm ay = axis_terms(&sax[GRID_N],     cy);
    const AxisTerm az = axis_terms(&sax[2 * GRID_N], cz);

    // flat index = ((ix*256)+iy)*256+iz ; fields are disjoint 8-bit lanes
    const unsigned xl = ax.il << 16, xr = ax.ir << 16;
    const unsigned yl = ay.il << 8,  yr = ay.ir << 8;
    const unsigned zl = az.il,       zr = az.ir;

    // 8 independent RT-cached gathers (values lives in L2: 64MB < 192MB)
    const float v000 = values[xl | yl | zl];
    const float v001 = values[xl | yl | zr];
    const float v010 = values[xl | yr | zl];
    const float v011 = values[xl | yr | zr];
    const float v100 = values[xr | yl | zl];
    const float v101 = values[xr | yl | zr];
    const float v110 = values[xr | yr | zl];
    const float v111 = values[xr | yr | zr];

    const float num =
          ax.wl * (ay.wl * (az.wl * v000 + az.wr * v001)
                 + ay.wr * (az.wl * v010 + az.wr * v011))
        + ax.wr * (ay.wl * (az.wl * v100 + az.wr * v101)
                 + ay.wr * (az.wl * v110 + az.wr * v111));

    const float den = ax.ov * ay.ov * az.ov;
    return num / den;
}

__global__ __launch_bounds__(256)
void RegularGridInterpolator_kernel(const float* __restrict__ x,
                                    const float* __restrict__ y,
                                    const float* __restrict__ z,
                                    const float* __restrict__ px,
                                    const float* __restrict__ py,
                                    const float* __restrict__ pz,
                                    const float* __restrict__ values,
                                    float* __restrict__ out,
                                    int n)
{
    __shared__ float sax[3 * GRID_N];   // px | py | pz, 3 KB

    const int t = threadIdx.x;

    // --- CDNA5 async tensor path: DMA the three 1KB axis arrays into LDS ---
    // Threads 0..63 each move one 16B chunk per axis (64 x 16B = 1KB per axis).
    // SGPR base per asm stmt is kernarg-uniform; VDST carries the LDS byte addr.
    if (t < 64) {
        const unsigned goff = (unsigned)t * 16u;
        const unsigned l0 = lds_byte_off(&sax[0]) + goff;
        asm volatile("global_load_async_to_lds_b128 %0, %1, %2 offset:0"
                     :: "v"(l0), "v"(goff),
                        "s"((unsigned long long)px) : "memory");
        const unsigned l1 = lds_byte_off(&sax[GRID_N]) + goff;
        asm volatile("global_load_async_to_lds_b128 %0, %1, %2 offset:0"
                     :: "v"(l1), "v"(goff),
                        "s"((unsigned long long)py) : "memory");
        const unsigned l2 = lds_byte_off(&sax[2 * GRID_N]) + goff;
        asm volatile("global_load_async_to_lds_b128 %0, %1, %2 offset:0"
                     :: "v"(l2), "v"(goff),
                        "s"((unsigned long long)pz) : "memory");
    }
    asm volatile("s_wait_asynccnt 0" ::: "memory");
    __syncthreads();

    // 2 consecutive points per iteration: b64 NT stream traffic and
    // 16 independent corner gathers in flight per thread.
    const int nthreads = (int)(blockDim.x * gridDim.x);
    const int stride2  = nthreads * 2;

    for (int i = ((int)(blockIdx.x * blockDim.x) + t) * 2; i < n; i += stride2) {
        if (i + 1 < n) {
            // NT loads: streaming coordinates must not evict the
            // L2-resident 64MB values grid. 8B-aligned (i is even).
            const f32x2 cx2 = __builtin_nontemporal_load((const f32x2*)&x[i]);
            const f32x2 cy2 = __builtin_nontemporal_load((const f32x2*)&y[i]);
            const f32x2 cz2 = __builtin_nontemporal_load((const f32x2*)&z[i]);

            f32x2 r;
            r.x = interp_one(sax, values, cx2.x, cy2.x, cz2.x);
            r.y = interp_one(sax, values, cx2.y, cy2.y, cz2.y);

            __builtin_nontemporal_store(r, (f32x2*)&out[i]);
        } else {
            const float cx = __builtin_nontemporal_load(&x[i]);
            const float cy = __builtin_nontemporal_load(&y[i]);
            const float cz = __builtin_nontemporal_load(&z[i]);
            __builtin_nontemporal_store(interp_one(sax, values, cx, cy, cz),
                                        &out[i]);
        }
    }
}

extern "C" void kernel_launch(void* const* d_in, const int* in_sizes, int n_in,
                              void* d_out, int out_size, void* d_ws, size_t ws_size,
                              hipStream_t stream)
{
    (void)n_in; (void)d_ws; (void)ws_size;
    const float* x      = (const float*)d_in[0];
    const float* y      = (const float*)d_in[1];
    const float* z      = (const float*)d_in[2];
    const float* px     = (const float*)d_in[3];
    const float* py     = (const float*)d_in[4];
    const float* pz     = (const float*)d_in[5];
    const float* values = (const float*)d_in[6];
    float* out = (float*)d_out;

    const int n = in_sizes[0];
    const int block = 256;
    // ~4 points per thread (2 grid-stride iterations of 2 points each).
    long long want = ((long long)n + (long long)block * 4 - 1) / ((long long)block * 4);
    if (want < 1) want = 1;
    if (want > 32768) want = 32768;
    const int blocks = (int)want;

    RegularGridInterpolator_kernel<<<blocks, block, 0, stream>>>(
        x, y, z, px, py, pz, values, out, n);
}